// SpeedSampler_56951266345227
// MI455X (gfx1250) — compile-verified
//
#include <hip/hip_runtime.h>

typedef __attribute__((ext_vector_type(2))) float v2f;
typedef __attribute__((ext_vector_type(8))) float v8f;

#define BFB 8
#define NQ 4096
#define CCH 64
#define QB 128            // queries per block
#define NTHREADS 128      // 4 waves (wave32)
#define CUR_STRIDE 68     // padded row stride (floats) for cur tile -> conflict-free A-frag reads
#define LOG_STRIDE 20     // padded row stride (floats) for logits -> conflict-free D-frag stores
#define PMAX 4
#define OUTC 66

// One deformable-attention stage, templated on point count P.
template<int P, bool WRITE_OUT>
__device__ __forceinline__ void run_stage(
    const float* __restrict__ we, const float* __restrict__ be,
    const float* __restrict__ ww, const float* __restrict__ bw,
    const float* __restrict__ value_b,            // pre_value base for this batch
    float* curbuf, float* Wlds, float* logits, int* metaIdx, float* metaW,
    float refx, float refy, float& sx, float& sy,
    int hsp, int wsp, float rx, float ry,
    float* __restrict__ outp)                     // out base for this block (when WRITE_OUT)
{
  const int tid  = threadIdx.x;
  const int lane = tid & 31;
  const int wave = tid >> 5;

  // ---- 1) pack [w_est | w_wt | 0] into a 64x16 LDS tile ----
  for (int i = tid; i < 64 * 16; i += NTHREADS) {
    int r = i >> 4, c = i & 15;
    float v = 0.f;
    if (c < 2 * P)       v = we[r * (2 * P) + c];
    else if (c < 3 * P)  v = ww[r * P + (c - 2 * P)];
    Wlds[i] = v;
  }
  __syncthreads();

  // ---- 2) WMMA projection: logits[16q x 16col] = cur[16x64] @ W[64x16] ----
  // A 16x4 f32 frag: lanes 0-15 hold K=k..k+1, lanes 16-31 hold K=k+2..k+3 (M = lane&15)
  // B 4x16 frag: same K split, N = lane&15.  D: VGPR j -> M=j (+8 upper half), N = lane&15.
  const int ncol = lane & 15;
  const int koff = (lane >> 4) << 1;   // 0 or 2
  for (int t = 0; t < 2; ++t) {
    const int qt = wave * 32 + t * 16;
    v8f acc = {};
#pragma unroll
    for (int k = 0; k < 16; ++k) {
      const int kk = 4 * k + koff;
      const float* arow = &curbuf[(qt + ncol) * CUR_STRIDE + kk];
      v2f a; a.x = arow[0];                a.y = arow[1];
      v2f b; b.x = Wlds[kk * 16 + ncol];   b.y = Wlds[(kk + 1) * 16 + ncol];
      acc = __builtin_amdgcn_wmma_f32_16x16x4_f32(false, a, false, b,
                                                  (short)0, acc, false, false);
    }
    const int rbase = qt + ((lane >> 4) << 3);
#pragma unroll
    for (int j = 0; j < 8; ++j)
      logits[(rbase + j) * LOG_STRIDE + ncol] = acc[j];
  }
  __syncthreads();

  // ---- 3) per-query scalar phase (lane <-> query) ----
  {
    const int ql = wave * 32 + lane;
    const float* L = &logits[ql * LOG_STRIDE];
    float delta[2 * P];
#pragma unroll
    for (int j = 0; j < 2 * P; ++j) delta[j] = L[j] + be[j];
    float wl[P];
#pragma unroll
    for (int j = 0; j < P; ++j) wl[j] = L[2 * P + j] + bw[j];
    // softmax over P
    float m = wl[0];
#pragma unroll
    for (int j = 1; j < P; ++j) m = fmaxf(m, wl[j]);
    float e[P]; float s = 0.f;
#pragma unroll
    for (int j = 0; j < P; ++j) { e[j] = __expf(wl[j] - m); s += e[j]; }
    const float inv = 1.f / s;
    // first-occurrence argmax (matches jnp.argmax)
    int mid = 0; float best = wl[0];
#pragma unroll
    for (int j = 1; j < P; ++j) if (wl[j] > best) { best = wl[j]; mid = j; }
    const float osx = sx, osy = sy;          // offsets use OLD speed_est
    sx += delta[2 * mid]; sy += delta[2 * mid + 1];
#pragma unroll
    for (int pt = 0; pt < P; ++pt) {
      const float off0 = osx + delta[2 * pt];
      const float off1 = osy + delta[2 * pt + 1];
      // x = (idx0 + 0.1*off0) * (w_sp/h_sp) - 0.5 ; y = (idx1 + 0.1*off1) * (h_sp/w_sp) - 0.5
      const float x = (refx + 0.1f * off0) * rx - 0.5f;
      const float y = (refy + 0.1f * off1) * ry - 0.5f;
      const float x0 = floorf(x), y0 = floorf(y);
      const float fx = x - x0, fy = y - y0;
      const int x0i = (int)x0, y0i = (int)y0;
      const float aw  = e[pt] * inv;
      const float w00 = aw * (1.f - fx) * (1.f - fy);
      const float w10 = aw * fx * (1.f - fy);
      const float w01 = aw * (1.f - fx) * fy;
      const float w11 = aw * fx * fy;
      const float wc[4] = {w00, w10, w01, w11};
#pragma unroll
      for (int c = 0; c < 4; ++c) {
        const int xi = x0i + (c & 1);
        const int yi = y0i + (c >> 1);
        const bool valid = (xi >= 0) && (xi < wsp) && (yi >= 0) && (yi < hsp);
        const int xc = xi < 0 ? 0 : (xi > wsp - 1 ? wsp - 1 : xi);
        const int yc = yi < 0 ? 0 : (yi > hsp - 1 ? hsp - 1 : yi);
        metaIdx[(ql * P + pt) * 4 + c] = yc * wsp + xc;
        metaW  [(ql * P + pt) * 4 + c] = valid ? wc[c] : 0.f;
      }
    }
  }
  __syncthreads();

  // ---- 4) gather phase: wave handles its 32 queries, 2 channels per lane ----
  {
    const int c0 = lane * 2;
    for (int q = 0; q < 32; ++q) {
      const int ql = wave * 32 + q;
      float a0 = 0.f, a1 = 0.f;
#pragma unroll
      for (int pt = 0; pt < P; ++pt) {
        const int*   mi = &metaIdx[(ql * P + pt) * 4];
        const float* mw = &metaW  [(ql * P + pt) * 4];
#pragma unroll
        for (int c = 0; c < 4; ++c) {
          const float2 v = *(const float2*)(value_b + (size_t)mi[c] * CCH + c0);
          const float w = mw[c];
          a0 = fmaf(w, v.x, a0);
          a1 = fmaf(w, v.y, a1);
        }
      }
      if (WRITE_OUT) {
        float* o = outp + (size_t)ql * OUTC + c0;
        o[0] = a0; o[1] = a1;
      } else {
        curbuf[ql * CUR_STRIDE + c0]     = a0;
        curbuf[ql * CUR_STRIDE + c0 + 1] = a1;
      }
    }
  }
  __syncthreads();
}

__global__ __launch_bounds__(NTHREADS)
void SpeedSampler_kernel(const float* __restrict__ cur_feat,
                         const float* __restrict__ cur_idx,
                         const float* __restrict__ pre_value,
                         const int* __restrict__ hsp_p,
                         const int* __restrict__ wsp_p,
                         const float* __restrict__ we1, const float* __restrict__ be1,
                         const float* __restrict__ ww1, const float* __restrict__ bw1,
                         const float* __restrict__ we2, const float* __restrict__ be2,
                         const float* __restrict__ ww2, const float* __restrict__ bw2,
                         const float* __restrict__ we3, const float* __restrict__ be3,
                         const float* __restrict__ ww3, const float* __restrict__ bw3,
                         float* __restrict__ out)
{
  __shared__ float curbuf[QB * CUR_STRIDE];
  __shared__ float Wlds[64 * 16];
  __shared__ float logits[QB * LOG_STRIDE];
  __shared__ int   metaIdx[QB * PMAX * 4];
  __shared__ float metaW[QB * PMAX * 4];

  const int hsp = *hsp_p, wsp = *wsp_p;
  const int HW  = hsp * wsp;
  const int tid  = threadIdx.x;
  const int lane = tid & 31;
  const int wave = tid >> 5;
  const int blocks_per_bf = NQ / QB;
  const int bf = blockIdx.x / blocks_per_bf;
  const int q0 = (blockIdx.x % blocks_per_bf) * QB;

  // cooperative, coalesced load of this block's cur tile into LDS
  {
    const float4* src = (const float4*)(cur_feat + ((size_t)bf * NQ + q0) * CCH);
    for (int i = tid; i < QB * (CCH / 4); i += NTHREADS) {
      const int q = i >> 4, ch = i & 15;
      const float4 v = src[q * 16 + ch];
      float* dst = &curbuf[q * CUR_STRIDE + ch * 4];
      dst[0] = v.x; dst[1] = v.y; dst[2] = v.z; dst[3] = v.w;
    }
  }
  __syncthreads();

  const int qg = q0 + wave * 32 + lane;
  const float refx = cur_idx[((size_t)bf * NQ + qg) * 2 + 0];
  const float refy = cur_idx[((size_t)bf * NQ + qg) * 2 + 1];
  const float* value_b = pre_value + (size_t)bf * HW * CCH;
  const float rx = (float)wsp / (float)hsp;
  const float ry = (float)hsp / (float)wsp;
  float* outp = out + ((size_t)bf * NQ + q0) * OUTC;
  float sx = 0.f, sy = 0.f;

  run_stage<4, false>(we1, be1, ww1, bw1, value_b, curbuf, Wlds, logits, metaIdx, metaW,
                      refx, refy, sx, sy, hsp, wsp, rx, ry, outp);
  run_stage<2, false>(we2, be2, ww2, bw2, value_b, curbuf, Wlds, logits, metaIdx, metaW,
                      refx, refy, sx, sy, hsp, wsp, rx, ry, outp);
  run_stage<1, true >(we3, be3, ww3, bw3, value_b, curbuf, Wlds, logits, metaIdx, metaW,
                      refx, refy, sx, sy, hsp, wsp, rx, ry, outp);

  // speed_est -> output cols 64,65
  {
    float* o = out + ((size_t)bf * NQ + qg) * OUTC;
    o[64] = sx; o[65] = sy;
  }
}

extern "C" void kernel_launch(void* const* d_in, const int* in_sizes, int n_in,
                              void* d_out, int out_size, void* d_ws, size_t ws_size,
                              hipStream_t stream) {
  const float* cur_feat = (const float*)d_in[0];
  const float* cur_idx  = (const float*)d_in[1];
  const float* pre_val  = (const float*)d_in[2];
  const int*   hsp      = (const int*)d_in[3];
  const int*   wsp      = (const int*)d_in[4];
  const float* we1 = (const float*)d_in[5];
  const float* be1 = (const float*)d_in[6];
  const float* ww1 = (const float*)d_in[7];
  const float* bw1 = (const float*)d_in[8];
  const float* we2 = (const float*)d_in[9];
  const float* be2 = (const float*)d_in[10];
  const float* ww2 = (const float*)d_in[11];
  const float* bw2 = (const float*)d_in[12];
  const float* we3 = (const float*)d_in[13];
  const float* be3 = (const float*)d_in[14];
  const float* ww3 = (const float*)d_in[15];
  const float* bw3 = (const float*)d_in[16];
  float* out = (float*)d_out;

  dim3 grid(BFB * NQ / QB), block(NTHREADS);
  hipLaunchKernelGGL(SpeedSampler_kernel, grid, block, 0, stream,
                     cur_feat, cur_idx, pre_val, hsp, wsp,
                     we1, be1, ww1, bw1,
                     we2, be2, ww2, bw2,
                     we3, be3, ww3, bw3,
                     out);
}